// max_extractor_59115929862504
// MI455X (gfx1250) — compile-verified
//
#include <hip/hip_runtime.h>
#include <stdint.h>

// Problem constants (match reference)
constexpr int kB = 8;
constexpr int kL = 2048;
constexpr int kD = 256;
constexpr int kT = 32;
constexpr int kSP = 8;    // para spans per batch
constexpr int kSA = 24;   // x / adu spans per batch
constexpr int kSS = 24;   // shell spans per batch

// Async-b128 pipeline constants (small spans)
constexpr int kChunkRows   = 4;                 // 4 rows * 1KB = 4KB = 256 thr * 16B
constexpr int kChunkFloats = kChunkRows * kD;   // 1024 floats
constexpr int kSlots       = 3;                 // ring: 2 in flight + 1 consuming

// TDM pipeline constants (para spans)
constexpr int kTdmRows        = 32;             // 32 rows * 1KB = 32KB per DMA
constexpr int kTdmChunkFloats = kTdmRows * kD;  // 8192 floats

// ---------------- TDM availability / arity ----------------
#if defined(__has_builtin)
#  if __has_builtin(__builtin_amdgcn_tensor_load_to_lds)
#    define HAVE_TDM 1
#  endif
#endif
#ifndef HAVE_TDM
#  define HAVE_TDM 0
#endif

typedef unsigned int v4u __attribute__((ext_vector_type(4)));
typedef int          v4i __attribute__((ext_vector_type(4)));
typedef int          v8i __attribute__((ext_vector_type(8)));

__device__ __forceinline__ int imin(int a, int b) { return a < b ? a : b; }

__device__ __forceinline__ void async_load_b128(uint32_t lds_byte_off, uint64_t gaddr) {
    // VDST = per-lane LDS byte address, VADDR = per-lane 64-bit global address (GV mode)
    asm volatile("global_load_async_to_lds_b128 %0, %1, off"
                 :: "v"(lds_byte_off), "v"(gaddr)
                 : "memory");
}

#if HAVE_TDM
// Issue one TDM 2D-tile load: rows x 256 f32, contiguous (stride0 = 256), into LDS.
// D# bit layout per CDNA5 ISA §8.3 (group0) / §8.4 (group1).
__device__ __forceinline__ void tdm_issue(const float* gsrc, uint32_t lds_byte, int rows) {
    const uint64_t ga = (uint64_t)gsrc;
    v4u g0;
    g0.x = 1u;                                                  // count=1, user mode
    g0.y = lds_byte;                                            // lds_addr (bytes)
    g0.z = (uint32_t)ga;                                        // global_addr[31:0]
    g0.w = (uint32_t)((ga >> 32) & 0x01FFFFFFu) | (2u << 30);   // addr[56:32] | type=2
    const uint32_t r16 = (uint32_t)rows & 0xffffu;
    v8i g1;
    g1[0] = (int)(2u << 16);          // wg_mask=0 | data_size=2 (4B) | no flags
    g1[1] = (int)(256u << 16);        // atomic_bar_addr=0 | tensor_dim0.lo16 = 256
    g1[2] = (int)(r16 << 16);         // tensor_dim0.hi16=0 | tensor_dim1.lo16 = rows
    g1[3] = (int)(256u << 16);        // tensor_dim1.hi16=0 | tile_dim0 = 256
    g1[4] = (int)r16;                 // tile_dim1 = rows | tile_dim2 = 0
    g1[5] = 256;                      // tensor_dim0_stride.lo32 = 256
    g1[6] = 0;                        // stride0.hi16 | stride1.lo16
    g1[7] = 0;                        // stride1.hi32
#  if __clang_major__ >= 23
    __builtin_amdgcn_tensor_load_to_lds(g0, g1, (v4i)0, (v4i)0, (v8i)0, 0);
#  else
    __builtin_amdgcn_tensor_load_to_lds(g0, g1, (v4i)0, (v4i)0, 0);
#  endif
}
#endif

// Triple-buffered (3-slot ring) async global->LDS max-pool over a contiguous block of rows.
// Each thread owns channel d = tid. Uses first kSlots*kChunkFloats floats of smem.
__device__ __forceinline__ float span_max_async(const float* __restrict__ rowbase,
                                                int nrows, float* smem, int tid) {
    float m = -3.402823466e38f;
    const int nchunks = nrows >> 2;
    if (nchunks > 0) {
        const uint64_t gbase = (uint64_t)rowbase + (uint64_t)tid * 16u;
        const uint32_t lbase = (uint32_t)(uintptr_t)&smem[0] + (uint32_t)tid * 16u;
        async_load_b128(lbase, gbase);
        if (nchunks > 1) async_load_b128(lbase + kChunkFloats * 4u, gbase + 4096u);
        for (int c = 0; c < nchunks; ++c) {
            if (c + 1 < nchunks) asm volatile("s_wait_asynccnt 1" ::: "memory");
            else                 asm volatile("s_wait_asynccnt 0" ::: "memory");
            __syncthreads();   // all waves' chunk-c data landed; prior consumes done
            if (c + 2 < nchunks) {
                const uint32_t slot = (uint32_t)((c + 2) % kSlots);
                async_load_b128(lbase + slot * (kChunkFloats * 4u),
                                gbase + (uint64_t)(c + 2) * 4096u);
            }
            const float* sp = &smem[(c % kSlots) * kChunkFloats + tid];
            #pragma unroll
            for (int r = 0; r < kChunkRows; ++r) m = fmaxf(m, sp[r * kD]);
        }
    }
    for (int r = nchunks * kChunkRows; r < nrows; ++r)   // tail rows (<4)
        m = fmaxf(m, rowbase[(size_t)r * kD + tid]);
    return m;
}

__global__ __launch_bounds__(256) void max_pool_kernel(
    const float* __restrict__ topic_reps,   // [B,T,D]
    const float* __restrict__ word_reps,    // [B,L,D]
    const int*   __restrict__ para_spans,   // [B,SP,3]
    const int*   __restrict__ x_spans,      // [B,SA,3]
    const int*   __restrict__ shell_spans,  // [B,SS,3]
    float*       __restrict__ out)          // topic | para | shell | x  (flat)
{
    __shared__ float smem[2 * kTdmChunkFloats];   // 64KB; async path uses first 12KB
    const int tid = threadIdx.x;
    const int blk = blockIdx.x;

    // ---------- topic max: blocks [0, kB) ----------
    if (blk < kB) {
        const float* base = topic_reps + (size_t)blk * kT * kD + tid;
        float m = -3.402823466e38f;
        #pragma unroll
        for (int t = 0; t < kT; ++t) m = fmaxf(m, base[(size_t)t * kD]);
        out[(size_t)blk * kD + tid] = m;
        return;
    }

    // ---------- span dispatch ----------
    const int sb = blk - kB;
    const bool is_para = (sb < kB * kSP);
    const int* span;
    size_t out_off;
    if (is_para) {                             // para -> out slot 1
        span = para_spans + (size_t)sb * 3;
        out_off = (size_t)kB * kD + (size_t)sb * kD;
    } else if (sb < kB * (kSP + kSS)) {        // shell -> span_reps (out slot 2)
        const int q = sb - kB * kSP;
        span = shell_spans + (size_t)q * 3;
        out_off = (size_t)kB * kD + (size_t)kB * kSP * kD + (size_t)q * kD;
    } else {                                   // x -> adu_reps (out slot 3)
        const int q = sb - kB * (kSP + kSS);
        span = x_spans + (size_t)q * 3;
        out_off = (size_t)kB * kD + (size_t)kB * (kSP + kSS) * kD + (size_t)q * kD;
    }

    const int elmo  = span[0];
    const int start = span[1];
    const int nrows = span[2] - start + 1;     // >= 1 by construction
    const float* rowbase = word_reps + ((size_t)elmo * kL + start) * kD;
    float m;

#if HAVE_TDM
    if (is_para) {
        // TDM double-buffered pipeline: one DMA per 32-row (32KB) chunk, wave 0 issues.
        m = -3.402823466e38f;
        const int nchunks = (nrows + kTdmRows - 1) / kTdmRows;
        const bool issuer = (tid < 32);        // wave 0 (wave32) — uniform per wave
        const uint32_t lds0 = (uint32_t)(uintptr_t)&smem[0];
        if (issuer) {
            tdm_issue(rowbase, lds0, imin(nrows, kTdmRows));
            if (nchunks > 1)
                tdm_issue(rowbase + (size_t)kTdmRows * kD,
                          lds0 + kTdmChunkFloats * 4u,
                          imin(nrows - kTdmRows, kTdmRows));
        }
        for (int c = 0; c < nchunks; ++c) {
            if (issuer) {                      // oldest DMA done (in-order per wave)
                if (c + 1 < nchunks) __builtin_amdgcn_s_wait_tensorcnt(1);
                else                 __builtin_amdgcn_s_wait_tensorcnt(0);
            }
            __syncthreads();                   // chunk c visible to all waves
            const int rowsC = imin(kTdmRows, nrows - c * kTdmRows);
            const float* sp = &smem[(c & 1) * kTdmChunkFloats + tid];
            for (int r = 0; r < rowsC; ++r) m = fmaxf(m, sp[r * kD]);
            __syncthreads();                   // all done reading slot (c&1)
            if (issuer && c + 2 < nchunks)     // refill the just-freed slot
                tdm_issue(rowbase + (size_t)(c + 2) * kTdmRows * kD,
                          lds0 + (uint32_t)((c & 1) * kTdmChunkFloats * 4),
                          imin(nrows - (c + 2) * kTdmRows, kTdmRows));
        }
    } else
#endif
    {
        m = span_max_async(rowbase, nrows, smem, tid);
    }

    out[out_off + tid] = m;
}

extern "C" void kernel_launch(void* const* d_in, const int* in_sizes, int n_in,
                              void* d_out, int out_size, void* d_ws, size_t ws_size,
                              hipStream_t stream) {
    const float* topic_reps  = (const float*)d_in[0];
    const float* word_reps   = (const float*)d_in[1];
    // d_in[2] = topic_lens (int64) — unused by the reference computation
    const int*   para_spans  = (const int*)d_in[3];
    const int*   x_spans     = (const int*)d_in[4];
    const int*   shell_spans = (const int*)d_in[5];
    float* out = (float*)d_out;

    const int nblocks = kB              // topic
                      + kB * kSP        // para
                      + kB * kSS        // shell (span_reps)
                      + kB * kSA;       // x (adu_reps)   = 456
    max_pool_kernel<<<dim3(nblocks), dim3(256), 0, stream>>>(
        topic_reps, word_reps, para_spans, x_spans, shell_spans, out);
}